// FrameHistoryRouter_7035156430879
// MI455X (gfx1250) — compile-verified
//
#include <hip/hip_runtime.h>
#include <math.h>

#define BATCH   2
#define DIM     192
#define HEIGHT  192
#define WIDTH   192
#define HW      (HEIGHT*WIDTH)        // 36864
#define HEADS   4
#define CH      (DIM/HEADS)           // 48
#define CACHE_C 192
#define KV_C    (CACHE_C + CH)        // 240
#define KEEP    192
#define C3      (3*DIM)               // 576
#define L2EPS   1e-12f
#define SPLITK  32
#define CHUNK   (HW/SPLITK)           // 1152
#define NTILE   4                     // 16x(16*NTILE) output tile per wave

typedef float v2f __attribute__((ext_vector_type(2)));
typedef float v8f __attribute__((ext_vector_type(8)));

// D = A(16x4 f32) * B(4x16 f32) + C(16x16 f32), wave32.
__device__ __forceinline__ v8f wmma4(v2f a, v2f b, v8f c) {
  return __builtin_amdgcn_wmma_f32_16x16x4_f32(false, a, false, b, (short)0, c,
                                               false, false);
}

// ---------------------------------------------------------------------------
// 1) Pointwise conv as GEMM: C[b,m,n] = sum_k A[m,k] * X[b,k,n] + bias[m]
//    One wave computes a 16(M) x 64(N) tile: A fragment reused for 4 WMMAs.
// ---------------------------------------------------------------------------
__global__ void gemm_pw_kernel(const float* __restrict__ A,     // [M,K]
                               const float* __restrict__ bias,  // [M]
                               const float* __restrict__ X,     // [BATCH][K,HW]
                               float* __restrict__ C,           // [BATCH][M,HW]
                               int M, int K) {
  const int lane = threadIdx.x & 31;
  const int wave = blockIdx.x * (blockDim.x >> 5) + (threadIdx.x >> 5);
  const int tilesM = M >> 4;
  const int tilesN = HW / (16 * NTILE);
  const int perBatch = tilesM * tilesN;
  const int b = wave / perBatch;
  if (b >= BATCH) return;                 // wave-uniform
  const int t = wave - b * perBatch;
  const int tm = t % tilesM;              // M fast -> waves in a block share X
  const int tn = t / tilesM;
  const int m0 = tm << 4;
  const int n0 = tn * (16 * NTILE);
  const int row  = lane & 15;
  const int koff = (lane >> 4) << 1;      // lanes 0-15: K+{0,1}; 16-31: K+{2,3}

  const float* Arow  = A + (size_t)(m0 + row) * K + koff;
  const float* Xbase = X + (size_t)b * K * HW + (size_t)koff * HW + n0 + row;

  v8f acc[NTILE] = {};
  for (int k = 0; k < K; k += 4) {
    const v2f a = *(const v2f*)(Arow + k);
    const float* x0 = Xbase + (size_t)k * HW;   // row k+koff
    const float* x1 = x0 + HW;                  // row k+koff+1
#pragma unroll
    for (int j = 0; j < NTILE; ++j) {
      v2f bf;
      bf.x = x0[j * 16];
      bf.y = x1[j * 16];
      acc[j] = wmma4(a, bf, acc[j]);
    }
  }
  // C/D layout: lane -> N = n0 + 16*j + (lane&15); VGPR r -> M = m0+r+8*(lane>=16)
  const int mbase = m0 + ((lane >> 4) << 3);
  float* Cbase = C + (size_t)b * M * HW + n0 + (lane & 15);
#pragma unroll
  for (int j = 0; j < NTILE; ++j) {
#pragma unroll
    for (int r = 0; r < 8; ++r) {
      const int m = mbase + r;
      Cbase[(size_t)m * HW + j * 16] = acc[j][r] + bias[m];
    }
  }
}

// ---------------------------------------------------------------------------
// 2) Depthwise 3x3 conv, SAME padding, + bias
// ---------------------------------------------------------------------------
__global__ void dwconv_kernel(const float* __restrict__ in,   // [BATCH][C3,HW]
                              const float* __restrict__ wdw,  // [C3,9]
                              const float* __restrict__ bdw,  // [C3]
                              float* __restrict__ out) {
  size_t idx = (size_t)blockIdx.x * blockDim.x + threadIdx.x;
  const size_t total = (size_t)BATCH * C3 * HW;
  if (idx >= total) return;
  const int n = (int)(idx % HW);
  const int o = (int)((idx / HW) % C3);
  const int y = n / WIDTH, x = n % WIDTH;
  const float* wp = wdw + o * 9;
  const float* ip = in + (idx - n);  // base of this (b,o) plane
  float s = bdw[o];
#pragma unroll
  for (int dy = -1; dy <= 1; ++dy) {
    const int yy = y + dy;
    if (yy < 0 || yy >= HEIGHT) continue;
#pragma unroll
    for (int dx = -1; dx <= 1; ++dx) {
      const int xx = x + dx;
      if (xx < 0 || xx >= WIDTH) continue;
      s += ip[yy * WIDTH + xx] * wp[(dy + 1) * 3 + (dx + 1)];
    }
  }
  out[idx] = s;
}

// ---------------------------------------------------------------------------
// 3) Row-wise L2 normalize (dim = HW) for q and k. One block per row.
// ---------------------------------------------------------------------------
__global__ void l2norm_kernel(const float* __restrict__ qkv,  // [BATCH][C3,HW]
                              float* __restrict__ qn,         // [BATCH][DIM,HW]
                              float* __restrict__ kn) {       // [BATCH][DIM,HW]
  __shared__ float red[256];
  const int rid   = blockIdx.x;               // 0 .. 2*BATCH*DIM-1
  const int which = rid / (BATCH * DIM);      // 0=q, 1=k
  const int r     = rid - which * BATCH * DIM;
  const int b     = r / DIM;
  const int o     = r - b * DIM;
  const float* src = qkv + ((size_t)b * C3 + which * DIM + o) * HW;
  float* dst = (which == 0 ? qn : kn) + ((size_t)b * DIM + o) * HW;

  const float4* s4 = (const float4*)src;
  float ss = 0.f;
  for (int i = threadIdx.x; i < HW / 4; i += blockDim.x) {
    const float4 v = s4[i];
    ss += v.x * v.x + v.y * v.y + v.z * v.z + v.w * v.w;
  }
  red[threadIdx.x] = ss;
  __syncthreads();
  for (int s = 128; s > 0; s >>= 1) {
    if (threadIdx.x < (unsigned)s) red[threadIdx.x] += red[threadIdx.x + s];
    __syncthreads();
  }
  const float inv = 1.0f / fmaxf(sqrtf(red[0]), L2EPS);
  float4* d4 = (float4*)dst;
  for (int i = threadIdx.x; i < HW / 4; i += blockDim.x) {
    float4 v = s4[i];
    v.x *= inv; v.y *= inv; v.z *= inv; v.w *= inv;
    d4[i] = v;
  }
}

__global__ void zero_kernel(float* __restrict__ p, int n) {
  const int i = blockIdx.x * blockDim.x + threadIdx.x;
  if (i < n) p[i] = 0.f;
}

// ---------------------------------------------------------------------------
// 4) attn += q @ k_cat^T  (split-K over HW, global f32 atomics)
//    out[c,d] = sum_n q[c,n] * kcat[d,n];  kcat = concat(k_cached, k_norm)
// ---------------------------------------------------------------------------
__global__ void attn_qk_kernel(const float* __restrict__ qn,
                               const float* __restrict__ kn,
                               const float* __restrict__ k_cached,
                               float* __restrict__ attn) { // [B*H][CH,KV_C]
  const int lane = threadIdx.x & 31;
  const int wave = blockIdx.x * (blockDim.x >> 5) + (threadIdx.x >> 5);
  const int tilesM = CH >> 4;    // 3
  const int tilesN = KV_C >> 4;  // 15
  const int perBH = tilesM * tilesN * SPLITK;
  const int bh = wave / perBH;
  if (bh >= BATCH * HEADS) return;        // wave-uniform
  int t = wave - bh * perBH;
  const int chunk = t / (tilesM * tilesN);
  t -= chunk * tilesM * tilesN;
  const int tm = t % tilesM;
  const int tn = t / tilesM;
  const int b = bh / HEADS;
  const int h = bh - b * HEADS;
  const int m0 = tm << 4;
  const int d0 = tn << 4;
  const int row  = lane & 15;
  const int koff = (lane >> 4) << 1;

  const float* qrow = qn + ((size_t)b * DIM + h * CH + m0 + row) * HW + koff;
  const int d = d0 + row;
  const float* krow = (d < CACHE_C)
      ? k_cached + (((size_t)b * HEADS + h) * CACHE_C + d) * HW + koff
      : kn + ((size_t)b * DIM + h * CH + (d - CACHE_C)) * HW + koff;

  const int nbeg = chunk * CHUNK;
  v8f acc = {};
  for (int n = 0; n < CHUNK; n += 4) {
    v2f a  = *(const v2f*)(qrow + nbeg + n);
    v2f bf = *(const v2f*)(krow + nbeg + n);
    acc = wmma4(a, bf, acc);
  }
  float* ap = attn + (size_t)bh * CH * KV_C + d0 + (lane & 15);
  const int mbase = m0 + ((lane >> 4) << 3);
#pragma unroll
  for (int r = 0; r < 8; ++r) {
    atomicAdd(ap + (size_t)(mbase + r) * KV_C, acc[r]);
  }
}

// ---------------------------------------------------------------------------
// 5) temperature-scaled softmax over KV_C=240. One block per row.
// ---------------------------------------------------------------------------
__global__ void softmax_kernel(const float* __restrict__ attn,
                               const float* __restrict__ temperature,
                               float* __restrict__ attn_sm) {
  __shared__ float red[256];
  const int rid = blockIdx.x;                 // BATCH*HEADS*CH rows
  const int h = (rid / CH) % HEADS;
  const float tmp = temperature[h];
  const float* src = attn + (size_t)rid * KV_C;
  float* dst = attn_sm + (size_t)rid * KV_C;
  const int i = threadIdx.x;
  const float v = (i < KV_C) ? src[i] * tmp : -3.4e38f;
  red[i] = v;
  __syncthreads();
  for (int s = 128; s > 0; s >>= 1) {
    if (i < s) red[i] = fmaxf(red[i], red[i + s]);
    __syncthreads();
  }
  const float m = red[0];
  __syncthreads();
  const float e = (i < KV_C) ? __expf(v - m) : 0.f;
  red[i] = e;
  __syncthreads();
  for (int s = 128; s > 0; s >>= 1) {
    if (i < s) red[i] += red[i + s];
    __syncthreads();
  }
  if (i < KV_C) dst[i] = e / red[0];
}

// ---------------------------------------------------------------------------
// 6) out2 = attn_sm @ v_cat ; v_cat = concat(v_cached, v_new) along rows
//    One wave computes 16(M) x 64(N); concat pointers hoisted per K-step.
// ---------------------------------------------------------------------------
__device__ __forceinline__ const float* vcat_row(const float* v_cached,
                                                 const float* qkv_dw,
                                                 int b, int h, int d) {
  if (d < CACHE_C)
    return v_cached + (((size_t)b * HEADS + h) * CACHE_C + d) * HW;
  return qkv_dw + ((size_t)b * C3 + 2 * DIM + h * CH + (d - CACHE_C)) * HW;
}

__global__ void attn_v_kernel(const float* __restrict__ attn_sm, // [B*H][CH,KV_C]
                              const float* __restrict__ qkv_dw,  // v at +2*DIM
                              const float* __restrict__ v_cached,
                              float* __restrict__ out2) {        // [BATCH][DIM,HW]
  const int lane = threadIdx.x & 31;
  const int wave = blockIdx.x * (blockDim.x >> 5) + (threadIdx.x >> 5);
  const int tilesM = CH >> 4;            // 3
  const int tilesN = HW / (16 * NTILE);  // 576
  const int perBH = tilesM * tilesN;
  const int bh = wave / perBH;
  if (bh >= BATCH * HEADS) return;        // wave-uniform
  const int t = wave - bh * perBH;
  const int tm = t % tilesM;
  const int tn = t / tilesM;
  const int b = bh / HEADS;
  const int h = bh - b * HEADS;
  const int m0 = tm << 4;
  const int n0 = tn * (16 * NTILE);
  const int row  = lane & 15;
  const int koff = (lane >> 4) << 1;

  const float* Arow = attn_sm + ((size_t)bh * CH + m0 + row) * KV_C + koff;
  v8f acc[NTILE] = {};
  for (int k = 0; k < KV_C; k += 4) {
    const v2f a = *(const v2f*)(Arow + k);
    const int ka = k + koff;
    const float* r0 = vcat_row(v_cached, qkv_dw, b, h, ka)     + n0 + row;
    const float* r1 = vcat_row(v_cached, qkv_dw, b, h, ka + 1) + n0 + row;
#pragma unroll
    for (int j = 0; j < NTILE; ++j) {
      v2f bf;
      bf.x = r0[j * 16];
      bf.y = r1[j * 16];
      acc[j] = wmma4(a, bf, acc[j]);
    }
  }
  float* Cbase = out2 + ((size_t)b * DIM + h * CH) * HW + n0 + (lane & 15);
  const int mbase = m0 + ((lane >> 4) << 3);
#pragma unroll
  for (int j = 0; j < NTILE; ++j) {
#pragma unroll
    for (int r = 0; r < 8; ++r) {
      Cbase[(size_t)(mbase + r) * HW + j * 16] = acc[j][r];
    }
  }
}

// ---------------------------------------------------------------------------
// 7) k_out/v_out = last KEEP rows of concat(cache, new), float4 copies
// ---------------------------------------------------------------------------
__global__ void kvout_kernel(const float* __restrict__ k_cached,
                             const float* __restrict__ v_cached,
                             const float* __restrict__ kn,
                             const float* __restrict__ qkv_dw,
                             float* __restrict__ k_out,
                             float* __restrict__ v_out) {
  const size_t hw4 = HW / 4;
  size_t idx = (size_t)blockIdx.x * blockDim.x + threadIdx.x;
  const size_t total = (size_t)BATCH * HEADS * KEEP * hw4;
  if (idx >= total) return;
  const size_t n4 = idx % hw4;
  size_t rest = idx / hw4;
  const int j = (int)(rest % KEEP);
  rest /= KEEP;
  const int h = (int)(rest % HEADS);
  const int b = (int)(rest / HEADS);
  const float4* ks;
  const float4* vs;
  if (j < CACHE_C - CH) {  // first 144 kept rows come from cache rows j+CH
    const size_t off = (((size_t)b * HEADS + h) * CACHE_C + j + CH) * hw4;
    ks = (const float4*)k_cached + off;
    vs = (const float4*)v_cached + off;
  } else {                 // last 48 rows are the new (normalized) k and raw v
    const int c = j - (CACHE_C - CH);
    ks = (const float4*)kn + ((size_t)b * DIM + h * CH + c) * hw4;
    vs = (const float4*)qkv_dw + ((size_t)b * C3 + 2 * DIM + h * CH + c) * hw4;
  }
  ((float4*)k_out)[idx] = ks[n4];
  ((float4*)v_out)[idx] = vs[n4];
}

// ---------------------------------------------------------------------------
extern "C" void kernel_launch(void* const* d_in, const int* in_sizes, int n_in,
                              void* d_out, int out_size, void* d_ws, size_t ws_size,
                              hipStream_t stream) {
  const float* x           = (const float*)d_in[0];
  const float* k_cached    = (const float*)d_in[1];
  const float* v_cached    = (const float*)d_in[2];
  const float* w_qkv       = (const float*)d_in[3];
  const float* b_qkv       = (const float*)d_in[4];
  const float* w_dw        = (const float*)d_in[5];
  const float* b_dw        = (const float*)d_in[6];
  const float* w_proj      = (const float*)d_in[7];
  const float* b_proj      = (const float*)d_in[8];
  const float* temperature = (const float*)d_in[9];

  float* out   = (float*)d_out;                                   // [B,DIM,HW]
  float* k_out = out + (size_t)BATCH * DIM * HW;                  // [B,H,KEEP,HW]
  float* v_out = k_out + (size_t)BATCH * HEADS * KEEP * HW;

  float* ws      = (float*)d_ws;
  float* qkv_pw  = ws;                                            // B*C3*HW
  float* qkv_dw  = qkv_pw + (size_t)BATCH * C3 * HW;              // B*C3*HW
  float* attn    = qkv_dw + (size_t)BATCH * C3 * HW;              // B*H*CH*KV_C
  float* attn_sm = attn + (size_t)BATCH * HEADS * CH * KV_C;
  // qkv_pw is dead after dwconv -> alias qn/kn/out2 into it (3*B*DIM == B*C3)
  float* qn   = qkv_pw;
  float* kn   = qkv_pw + (size_t)BATCH * DIM * HW;
  float* out2 = qkv_pw + 2 * (size_t)BATCH * DIM * HW;

  const int THREADS = 256;  // 8 wave32s

  // 1) qkv = w_qkv @ x + b_qkv           (WMMA f32, 16x64 tiles)
  {
    const int waves = BATCH * (C3 / 16) * (HW / (16 * NTILE));
    gemm_pw_kernel<<<waves / 8, THREADS, 0, stream>>>(w_qkv, b_qkv, x, qkv_pw,
                                                      C3, DIM);
  }
  // 2) depthwise 3x3 + b_dw
  {
    const size_t total = (size_t)BATCH * C3 * HW;
    dwconv_kernel<<<(unsigned)((total + THREADS - 1) / THREADS), THREADS, 0,
                    stream>>>(qkv_pw, w_dw, b_dw, qkv_dw);
  }
  // 3) L2 normalize q, k over HW
  l2norm_kernel<<<2 * BATCH * DIM, THREADS, 0, stream>>>(qkv_dw, qn, kn);
  // 4) zero attn accumulator
  {
    const int n = BATCH * HEADS * CH * KV_C;
    zero_kernel<<<(n + THREADS - 1) / THREADS, THREADS, 0, stream>>>(attn, n);
  }
  // 5) attn += q @ kcat^T (split-K x32, WMMA f32 + f32 atomics)
  {
    const int waves = BATCH * HEADS * (CH / 16) * (KV_C / 16) * SPLITK;
    attn_qk_kernel<<<waves / 8, THREADS, 0, stream>>>(qn, kn, k_cached, attn);
  }
  // 6) softmax(temperature * attn)
  softmax_kernel<<<BATCH * HEADS * CH, THREADS, 0, stream>>>(attn, temperature,
                                                             attn_sm);
  // 7) out2 = attn_sm @ vcat             (WMMA f32, 16x64 tiles)
  {
    const int waves = BATCH * HEADS * (CH / 16) * (HW / (16 * NTILE));
    attn_v_kernel<<<waves / 8, THREADS, 0, stream>>>(attn_sm, qkv_dw, v_cached,
                                                     out2);
  }
  // 8) out = w_proj @ out2 + b_proj      (WMMA f32, 16x64 tiles) -> d_out
  {
    const int waves = BATCH * (DIM / 16) * (HW / (16 * NTILE));
    gemm_pw_kernel<<<waves / 8, THREADS, 0, stream>>>(w_proj, b_proj, out2, out,
                                                      DIM, DIM);
  }
  // 9) k_out / v_out cache rotation (pure bandwidth, float4)
  {
    const size_t total = (size_t)BATCH * HEADS * KEEP * (HW / 4);
    kvout_kernel<<<(unsigned)((total + THREADS - 1) / THREADS), THREADS, 0,
                   stream>>>(k_cached, v_cached, kn, qkv_dw, k_out, v_out);
  }
}